// GCN_2L_Model_55594056680044
// MI455X (gfx1250) — compile-verified
//
#include <hip/hip_runtime.h>
#include <hip/hip_bf16.h>

typedef __attribute__((ext_vector_type(2))) float v2f;
typedef __attribute__((ext_vector_type(8))) float v8f;

#define FEATS 64

// ---------- zero workspace ----------
__global__ void zero_f32(float* __restrict__ p, long long n) {
  long long i = (long long)blockIdx.x * blockDim.x + threadIdx.x;
  if (i < n) p[i] = 0.0f;
}

// ---------- degree counting (exact in f32: degrees << 2^24) ----------
__global__ void degree_kernel(const int* __restrict__ src, const int* __restrict__ dst,
                              float* __restrict__ deg_out, float* __restrict__ deg_in, int ne) {
  int e = blockIdx.x * blockDim.x + threadIdx.x;
  if (e < ne) {
    atomicAdd(&deg_out[src[e]], 1.0f);
    atomicAdd(&deg_in[dst[e]], 1.0f);
  }
}

// ---------- deg -> norm (in place): where(deg>0, deg, 1)^-0.5 ----------
__global__ void norm_kernel(float* __restrict__ deg_out, float* __restrict__ deg_in, int n) {
  int i = blockIdx.x * blockDim.x + threadIdx.x;
  if (i < n) {
    float d0 = deg_out[i];
    deg_out[i] = (d0 > 0.0f) ? (1.0f / sqrtf(d0)) : 1.0f;
    float d1 = deg_in[i];
    deg_in[i] = (d1 > 0.0f) ? (1.0f / sqrtf(d1)) : 1.0f;
  }
}

// ---------- layer-1 scatter: agg[dst] += x[src] * norm_out[src], per (edge,feat) ----------
__global__ void scatter_feat_kernel(const float* __restrict__ x, const float* __restrict__ norm_out,
                                    const int* __restrict__ src, const int* __restrict__ dst,
                                    float* __restrict__ agg, long long total) {
  long long t = (long long)blockIdx.x * blockDim.x + threadIdx.x;
  if (t >= total) return;
  int e = (int)(t >> 6);
  int f = (int)(t & 63);
  int s = src[e];
  int d = dst[e];
  float v = x[(size_t)s * FEATS + f] * norm_out[s];
  atomicAdd(&agg[(size_t)d * FEATS + f], v);
}

// ---------- h1 = relu((agg * norm_in) @ W1 + b1), in place over agg, f32 WMMA ----------
// One wave per 16 output rows. A: 16x4 f32 fragments per ISA layout
// (lanes 0-15: K=0,1 in v0,v1; lanes 16-31: K=2,3). B mirrors with N across lanes.
// C/D: lanes 0-15 -> N=lane, M=0..7 in c[0..7]; lanes 16-31 -> N=lane-16, M=8..15.
__global__ void __launch_bounds__(32) gemm_bias_relu_wmma(
    float* __restrict__ h,            // [n,64] in: agg, out: relu result (in place)
    const float* __restrict__ norm_in,
    const float* __restrict__ W1,     // [64,64] row-major
    const float* __restrict__ b1,     // [64]
    int n) {
  const int lane = threadIdx.x;      // 0..31
  const int half = lane >> 4;        // 0 or 1
  const int l16  = lane & 15;

  int rowA = blockIdx.x * 16 + l16;
  int rowC = (rowA < n) ? rowA : (n - 1);          // clamp (intra-block only; safe in place)
  const float scale = norm_in[rowC];
  const float* arow = h + (size_t)rowC * FEATS;

  // Preload all 16 K-chunk A-fragments, scaled by norm_in[row].
  v2f afrag[16];
#pragma unroll
  for (int k = 0; k < 16; ++k) {
    int kk = k * 4 + half * 2;
    v2f a;
    a.x = arow[kk]     * scale;
    a.y = arow[kk + 1] * scale;
    afrag[k] = a;
  }

  const int rowBase = blockIdx.x * 16 + half * 8;  // D-matrix row origin for this lane half
#pragma unroll
  for (int nt = 0; nt < 4; ++nt) {
    const int col = nt * 16 + l16;
    v8f c = {};
#pragma unroll
    for (int k = 0; k < 16; ++k) {
      int kk = k * 4 + half * 2;
      v2f bf;
      bf.x = W1[(size_t)kk * FEATS + col];
      bf.y = W1[(size_t)(kk + 1) * FEATS + col];
      // 8 args: (neg_a, A, neg_b, B, c_mod, C, reuse_a, reuse_b)
      c = __builtin_amdgcn_wmma_f32_16x16x4_f32(false, afrag[k], false, bf,
                                                (short)0, c, false, false);
    }
    const float bias = b1[col];
#pragma unroll
    for (int r = 0; r < 8; ++r) {
      int row = rowBase + r;
      if (row < n) {
        float v = c[r] + bias;
        h[(size_t)row * FEATS + col] = (v > 0.0f) ? v : 0.0f;
      }
    }
  }
}

// ---------- layer-2 projection: s[i] = (h1[i] . W2) * norm_out[i] ----------
__global__ void node_project_kernel(const float* __restrict__ h1, const float* __restrict__ norm_out,
                                    const float* __restrict__ W2, float* __restrict__ s, int n) {
  int i = blockIdx.x * blockDim.x + threadIdx.x;
  if (i >= n) return;
  const float4* r = (const float4*)(h1 + (size_t)i * FEATS);
  const float4* w = (const float4*)W2;
  float acc = 0.0f;
#pragma unroll
  for (int k = 0; k < FEATS / 4; ++k) {
    float4 a = r[k];
    float4 b = w[k];
    acc += a.x * b.x + a.y * b.y + a.z * b.z + a.w * b.w;
  }
  s[i] = acc * norm_out[i];
}

// ---------- layer-2 scatter: z[dst] += s[src] ----------
__global__ void scatter_scalar_kernel(const float* __restrict__ s, const int* __restrict__ src,
                                      const int* __restrict__ dst, float* __restrict__ z, int ne) {
  int e = blockIdx.x * blockDim.x + threadIdx.x;
  if (e < ne) atomicAdd(&z[dst[e]], s[src[e]]);
}

// ---------- out = sigmoid(z * norm_in + b2) ----------
__global__ void sigmoid_kernel(const float* __restrict__ z, const float* __restrict__ norm_in,
                               const float* __restrict__ b2, float* __restrict__ out, int n) {
  int i = blockIdx.x * blockDim.x + threadIdx.x;
  if (i < n) {
    float t = z[i] * norm_in[i] + b2[0];
    out[i] = 1.0f / (1.0f + expf(-t));
  }
}

extern "C" void kernel_launch(void* const* d_in, const int* in_sizes, int n_in,
                              void* d_out, int out_size, void* d_ws, size_t ws_size,
                              hipStream_t stream) {
  const float* x  = (const float*)d_in[0];
  const int* src  = (const int*)d_in[1];
  const int* dst  = (const int*)d_in[2];
  const float* W1 = (const float*)d_in[3];
  const float* b1 = (const float*)d_in[4];
  const float* W2 = (const float*)d_in[5];
  const float* b2 = (const float*)d_in[6];
  float* out = (float*)d_out;

  const int n  = in_sizes[0] / FEATS;   // nodes
  const int ne = in_sizes[1];           // edges

  // Workspace layout (floats): norm_out[n] | norm_in[n] | agg/h1[n*64] | s[n] | z2[n]
  float* ws       = (float*)d_ws;
  float* norm_out = ws;
  float* norm_in  = ws + (size_t)n;
  float* agg      = ws + (size_t)2 * n;
  float* svec     = agg + (size_t)n * FEATS;
  float* z2       = svec + (size_t)n;

  const long long total_ws = (long long)n * (FEATS + 4);

  {
    long long blocks = (total_ws + 255) / 256;
    zero_f32<<<(unsigned)blocks, 256, 0, stream>>>(ws, total_ws);
  }
  degree_kernel<<<(ne + 255) / 256, 256, 0, stream>>>(src, dst, norm_out, norm_in, ne);
  norm_kernel<<<(n + 255) / 256, 256, 0, stream>>>(norm_out, norm_in, n);
  {
    long long total = (long long)ne * FEATS;
    long long blocks = (total + 255) / 256;
    scatter_feat_kernel<<<(unsigned)blocks, 256, 0, stream>>>(x, norm_out, src, dst, agg, total);
  }
  gemm_bias_relu_wmma<<<(n + 15) / 16, 32, 0, stream>>>(agg, norm_in, W1, b1, n);
  node_project_kernel<<<(n + 255) / 256, 256, 0, stream>>>(agg, norm_out, W2, svec, n);
  scatter_scalar_kernel<<<(ne + 255) / 256, 256, 0, stream>>>(svec, src, dst, z2, ne);
  sigmoid_kernel<<<(n + 255) / 256, 256, 0, stream>>>(z2, norm_in, b2, out, n);
}